// Finetune_61658550501869
// MI455X (gfx1250) — compile-verified
//
#include <hip/hip_runtime.h>
#include <stdint.h>

// Problem constants (fixed by the reference: INTERVAL=4 -> Q=16, L=17)
#define LUT_L   17
#define LUT_LL  289          // 17*17
#define LUT_N   14739        // 17^3 * 3 floats (~59 KB in LDS)
#define LUT_N4  3684         // floor(LUT_N/4) float4 async copies (+3 tail B32)
#define IMG_HW  2073600      // 1080*1920
#define HW4     518400       // IMG_HW/4
#define NGRP    1036800      // NPIX/4 float4 pixel groups
#define TPB     256
#define BLOCKS  810
#define ITERS   5            // NGRP / (TPB*BLOCKS) == 5 exactly -> EXEC stays full

typedef __attribute__((ext_vector_type(2))) float v2f;   // WMMA 16x16x4 f32 A/B operand
typedef __attribute__((ext_vector_type(4))) float v4f;   // B128 image vectors
typedef __attribute__((ext_vector_type(8))) float v8f;   // WMMA 16x16 f32 C/D accumulator

__device__ __forceinline__ float bperm(float v, int srcLane) {
  // wave32 backward permute: dst[lane] = src[srcLane]
  return __int_as_float(__builtin_amdgcn_ds_bpermute(srcLane * 4, __float_as_int(v)));
}

__global__ __launch_bounds__(TPB) void lut3d_ccm_kernel(
    const float* __restrict__ img, const float* __restrict__ lut,
    const float* __restrict__ ccm, float* __restrict__ out) {
  __shared__ float sh[LUT_N];
  const int tid = threadIdx.x;

  // ---- Stage raw LUT into LDS with CDNA5 async global->LDS B128 copies (ASYNCcnt) ----
  {
    uint32_t lds0 = (uint32_t)(uintptr_t)(&sh[0]);              // low 32 bits = LDS byte offset
    unsigned long long gbase = (unsigned long long)(uintptr_t)lut;
    for (int i = tid; i < LUT_N4; i += TPB) {
      uint32_t ldsaddr = lds0 + (uint32_t)(i * 16);
      uint32_t voff = (uint32_t)(i * 16);
      asm volatile("global_load_async_to_lds_b128 %0, %1, %2"
                   :: "v"(ldsaddr), "v"(voff), "s"(gbase) : "memory");
    }
    if (tid < 3) {                                              // 14739 = 3684*4 + 3 tail
      uint32_t ldsaddr = lds0 + (uint32_t)((LUT_N4 * 16) + tid * 4);
      uint32_t voff = (uint32_t)((LUT_N4 * 16) + tid * 4);
      asm volatile("global_load_async_to_lds_b32 %0, %1, %2"
                   :: "v"(ldsaddr), "v"(voff), "s"(gbase) : "memory");
    }
    asm volatile("s_wait_asynccnt 0" ::: "memory");
  }
  __syncthreads();
  // Quantize LUT in place: w = clamp(round(x*127), -127, 127)  (RNE)
  for (int i = tid; i < LUT_N; i += TPB) {
    float x = sh[i];
    sh[i] = fminf(fmaxf(rintf(x * 127.0f), -127.0f), 127.0f);
  }
  __syncthreads();

  // ---- WMMA A operand = CCM (16x4 f32 A layout: lanes0-15 K=0/1, lanes16-31 K=2/3) ----
  // out[m] = sum_k ccm[m][k] * pix[k]  ->  A[m][k] = ccm[m][k], rows m>=3 zero.
  const int lane = tid & 31;
  const int m = lane & 15;
  v2f amat; amat.x = 0.0f; amat.y = 0.0f;
  if (m < 3) {
    if (lane < 16) { amat.x = ccm[m * 3 + 0]; amat.y = ccm[m * 3 + 1]; }   // K=0, K=1
    else           { amat.x = ccm[m * 3 + 2]; }                            // K=2 (K=3 = 0)
  }
  const v8f czero = {};
  // Branchless half-wave select masks (force v_cndmask/mul, never EXEC branches)
  const float selLo = (lane < 16) ? 1.0f : 0.0f;
  const float selHi = 1.0f - selLo;
  const int loLane = lane & 15;

  const int gbase = blockIdx.x * TPB + tid;
  #pragma unroll 1
  for (int it = 0; it < ITERS; ++it) {
    const int gi = gbase + it * (TPB * BLOCKS);   // float4-group id, exact coverage
    const int bimg = gi / HW4;
    const int p4 = gi - bimg * HW4;
    const int base3 = bimg * 3 * IMG_HW + p4 * 4;

    // Streaming B128 non-temporal loads: don't pollute L2 (keep LUT resident)
    const v4f r4 = __builtin_nontemporal_load((const v4f*)(img + base3));
    const v4f g4 = __builtin_nontemporal_load((const v4f*)(img + base3 + IMG_HW));
    const v4f b4 = __builtin_nontemporal_load((const v4f*)(img + base3 + 2 * IMG_HW));

    v4f o0v, o1v, o2v;
    #pragma unroll
    for (int j = 0; j < 4; ++j) {
      const float r = r4[j], g = g4[j], b = b4[j];

      int r1 = (int)floorf(r * 0.0625f);
      int g1 = (int)floorf(g * 0.0625f);
      int b1 = (int)floorf(b * 0.0625f);
      const float fr = r * 0.0625f - (float)r1;
      const float fg = g * 0.0625f - (float)g1;
      const float fb = b * 0.0625f - (float)b1;
      const int r2 = min(r1 + 1, LUT_L - 1);
      const int g2 = min(g1 + 1, LUT_L - 1);
      const int b2 = min(b1 + 1, LUT_L - 1);

      const int ir1 = r1 * LUT_LL, ir2 = r2 * LUT_LL;
      const int jg1 = g1 * LUT_L,  jg2 = g2 * LUT_L;
      const int i000 = (ir1 + jg1 + b1) * 3, i001 = (ir1 + jg1 + b2) * 3;
      const int i010 = (ir1 + jg2 + b1) * 3, i011 = (ir1 + jg2 + b2) * 3;
      const int i100 = (ir2 + jg1 + b1) * 3, i101 = (ir2 + jg1 + b2) * 3;
      const int i110 = (ir2 + jg2 + b1) * 3, i111 = (ir2 + jg2 + b2) * 3;

      float ox = 0.f, oy = 0.f, oz = 0.f;
      #pragma unroll
      for (int ch = 0; ch < 3; ++ch) {
        const float p000 = sh[i000 + ch], p001 = sh[i001 + ch];
        const float p010 = sh[i010 + ch], p011 = sh[i011 + ch];
        const float p100 = sh[i100 + ch], p101 = sh[i101 + ch];
        const float p110 = sh[i110 + ch], p111 = sh[i111 + ch];
        const float c00 = p000 + (p100 - p000) * fr;
        const float c01 = p001 + (p101 - p001) * fr;
        const float c10 = p010 + (p110 - p010) * fr;
        const float c11 = p011 + (p111 - p011) * fr;
        const float c0 = c00 + (c10 - c00) * fg;
        const float c1 = c01 + (c11 - c01) * fg;
        const float o = c0 + (c1 - c0) * fb;
        if (ch == 0) ox = o; else if (ch == 1) oy = o; else oz = o;
      }

      // ---- CCM via two V_WMMA_F32_16X16X4_F32 covering the wave's 32 sub-pixels ----
      // B (4x16 f32): VGPR0 lanes0-15 = K=0 row, lanes16-31 = K=2; VGPR1 = K=1 / K=3.
      v2f bv1, bv2;
      const float blow = bperm(oz, loLane);               // pixel (lane&15)'s b
      bv1.x = (lane < 16) ? ox : blow;                    // K0=r(n) / K2=b(n)
      bv1.y = (lane < 16) ? oy : 0.0f;                    // K1=g(n) / K3=0
      const float rhi = bperm(ox, (lane + 16) & 31);      // pixel (lane+16)'s r
      const float ghi = bperm(oy, (lane + 16) & 31);      // pixel (lane+16)'s g
      bv2.x = (lane < 16) ? rhi : oz;                     // upper half: own pixel's b (K=2)
      bv2.y = (lane < 16) ? ghi : 0.0f;

      const v8f d1 = __builtin_amdgcn_wmma_f32_16x16x4_f32(
          false, amat, false, bv1, (short)0, czero, false, false);
      const v8f d2 = __builtin_amdgcn_wmma_f32_16x16x4_f32(
          false, amat, false, bv2, (short)0, czero, false, false);

      // D layout: VGPR c holds M=c (lanes0-15, N=lane). Channel c of pixel n -> d[c] @ lane n.
      // Branchless blend so ds_bpermute runs with full EXEC (no saveexec predication).
      const float bp0 = bperm(d2[0], loLane);
      const float bp1 = bperm(d2[1], loLane);
      const float bp2 = bperm(d2[2], loLane);
      float o0 = d1[0] * selLo + bp0 * selHi;
      float o1 = d1[1] * selLo + bp1 * selHi;
      float o2 = d1[2] * selLo + bp2 * selHi;

      o0v[j] = fminf(fmaxf(o0, 1e-8f), 1.0f);
      o1v[j] = fminf(fmaxf(o1, 1e-8f), 1.0f);
      o2v[j] = fminf(fmaxf(o2, 1e-8f), 1.0f);
    }

    // Streaming B128 non-temporal stores
    __builtin_nontemporal_store(o0v, (v4f*)(out + base3));
    __builtin_nontemporal_store(o1v, (v4f*)(out + base3 + IMG_HW));
    __builtin_nontemporal_store(o2v, (v4f*)(out + base3 + 2 * IMG_HW));
  }
}

extern "C" void kernel_launch(void* const* d_in, const int* in_sizes, int n_in,
                              void* d_out, int out_size, void* d_ws, size_t ws_size,
                              hipStream_t stream) {
  (void)in_sizes; (void)n_in; (void)out_size; (void)d_ws; (void)ws_size;
  const float* img = (const float*)d_in[0];   // (2,3,1080,1920) f32
  const float* lut = (const float*)d_in[1];   // (17^3, 3) f32 (trailing 1,1 dims)
  const float* ccm = (const float*)d_in[2];   // (3,3) f32
  float* out = (float*)d_out;                 // (2,3,1080,1920) f32
  hipLaunchKernelGGL(lut3d_ccm_kernel, dim3(BLOCKS), dim3(TPB), 0, stream,
                     img, lut, ccm, out);
}